// VecDGCNN_67508295958613
// MI455X (gfx1250) — compile-verified
//
#include <hip/hip_runtime.h>

typedef __attribute__((ext_vector_type(2))) float v2f;
typedef __attribute__((ext_vector_type(8))) float v8f;

#define NPTS 1024
#define KNN  16
#define CCAT 352
#define VEPS 1e-12f
#define SLOPE 0.2f

// Wait for all outstanding async (to-LDS) transfers.
__device__ __forceinline__ void wait_async0() {
#if __has_builtin(__builtin_amdgcn_s_wait_asynccnt)
    __builtin_amdgcn_s_wait_asynccnt(0);
#else
    asm volatile("s_wait_asynccnt 0x0" ::: "memory");
#endif
}

// ---------------------------------------------------------------------------
// Vector-neuron LeakyReLU on a single (x, direction) pair of 3-vectors.
__device__ __forceinline__ void vnact3(const float xv[3], const float dv[3], float out[3]) {
    float dn  = sqrtf(dv[0]*dv[0] + dv[1]*dv[1] + dv[2]*dv[2]);
    float inv = 1.0f / fmaxf(dn, VEPS);
    float k0 = dv[0]*inv, k1 = dv[1]*inv, k2 = dv[2]*inv;
    float dot = xv[0]*k0 + xv[1]*k1 + xv[2]*k2;
    float act = (dot >= 0.0f) ? dot : SLOPE * dot;
    float t = act - dot;
    out[0] = xv[0] + t*k0;
    out[1] = xv[1] + t*k1;
    out[2] = xv[2] + t*k2;
}

// ---------------------------------------------------------------------------
// KNN: one block per batch; points staged in LDS; each thread keeps a sorted
// top-16 (smallest squared distance, self included like top_k(-d2)).
__global__ __launch_bounds__(256) void knn_kernel(const float* __restrict__ x,
                                                  int* __restrict__ idx) {
    __shared__ float px[NPTS], py[NPTS], pz[NPTS];
    const int b = blockIdx.x;
    for (int i = threadIdx.x; i < NPTS; i += 256) {
        px[i] = x[(b*3 + 0)*NPTS + i];
        py[i] = x[(b*3 + 1)*NPTS + i];
        pz[i] = x[(b*3 + 2)*NPTS + i];
    }
    __syncthreads();
    for (int n = threadIdx.x; n < NPTS; n += 256) {
        float bd[KNN];
        int   bi[KNN];
        #pragma unroll
        for (int k = 0; k < KNN; ++k) { bd[k] = 3.4e38f; bi[k] = 0; }
        const float ax = px[n], ay = py[n], az = pz[n];
        for (int j = 0; j < NPTS; ++j) {
            float dx = px[j]-ax, dy = py[j]-ay, dz = pz[j]-az;
            float d2 = dx*dx + dy*dy + dz*dz;
            if (d2 < bd[KNN-1]) {
                int pos = KNN-1;
                while (pos > 0 && bd[pos-1] > d2) {
                    bd[pos] = bd[pos-1]; bi[pos] = bi[pos-1]; --pos;
                }
                bd[pos] = d2; bi[pos] = j;
            }
        }
        for (int k = 0; k < KNN; ++k) idx[(b*NPTS + n)*KNN + k] = bi[k];
    }
}

__global__ void zero_kernel(float* p, int n) {
    int i = blockIdx.x * blockDim.x + threadIdx.x;
    if (i < n) p[i] = 0.0f;
}

// ---------------------------------------------------------------------------
// EdgeConv + VN-LeakyReLU + mean-over-K.  One block = one (batch, point).
// GEMM1: W(Cout x 2Cin) @ E(2Cin x 48), GEMM2: Wd(Cout x Cout) @ D, both via
// V_WMMA_F32_16X16X4_F32. Columns: col = comp*16 + k.
__global__ __launch_bounds__(128) void edgeconv_kernel(
    const float* __restrict__ Xin, int cstrideIn, int coffIn, int Cin,
    const float* __restrict__ W, const float* __restrict__ Wd, int Cout,
    const int* __restrict__ idx, float* __restrict__ Xout, int coffOut) {

    __shared__ float Es[64][48];     // K-tile of edge features
    __shared__ float Dbuf[128][48];  // GEMM1 output
    __shared__ float D2buf[128][48]; // GEMM2 output (directions)
    __shared__ int   idxs[KNN];

    const int n   = blockIdx.x;
    const int b   = blockIdx.y;
    const int tid = threadIdx.x;
    const int wave  = tid >> 5;
    const int lane  = tid & 31;
    const int lhalf = lane >> 4;     // 0: lanes 0-15, 1: lanes 16-31
    const int l15   = lane & 15;
    const int K2    = 2 * Cin;
    const int nRT   = Cout >> 4;     // 16-row tiles of Cout

    if (tid < KNN) idxs[tid] = idx[(b*NPTS + n)*KNN + tid];

    v8f acc[2][3];
    #pragma unroll
    for (int i = 0; i < 2; ++i)
        #pragma unroll
        for (int ct = 0; ct < 3; ++ct) acc[i][ct] = (v8f){};

    __builtin_prefetch(W, 0, 1);
    const int nkt = (K2 + 63) >> 6;
    for (int kt = 0; kt < nkt; ++kt) {
        __syncthreads();
        // stage 64 x 48 edge-feature tile
        for (int e = tid; e < 64*48; e += 128) {
            int r = e / 48, cidx = e - r*48;
            int d = cidx >> 4, k = cidx & 15;
            int g = kt*64 + r;
            float v = 0.0f;
            if (g < Cin) {
                int base = ((b*cstrideIn + coffIn + g)*3 + d)*NPTS;
                v = Xin[base + idxs[k]] - Xin[base + n];
            } else if (g < K2) {
                v = Xin[((b*cstrideIn + coffIn + (g - Cin))*3 + d)*NPTS + n];
            }
            Es[r][cidx] = v;
        }
        __syncthreads();
        for (int s = 0; s < 16; ++s) {
            int kg   = kt*64 + s*4 + lhalf*2;  // global K index of this lane's pair
            int rloc = s*4 + lhalf*2;
            v2f bf[3];
            #pragma unroll
            for (int ct = 0; ct < 3; ++ct) {
                bf[ct].x = Es[rloc][ct*16 + l15];
                bf[ct].y = Es[rloc+1][ct*16 + l15];
            }
            #pragma unroll
            for (int i = 0; i < 2; ++i) {
                int rt = wave + i*4;           // wave-uniform -> EXEC stays full
                if (rt < nRT) {
                    int row = rt*16 + l15;
                    v2f af;
                    af.x = (kg     < K2) ? W[row*K2 + kg]     : 0.0f;
                    af.y = (kg + 1 < K2) ? W[row*K2 + kg + 1] : 0.0f;
                    #pragma unroll
                    for (int ct = 0; ct < 3; ++ct)
                        acc[i][ct] = __builtin_amdgcn_wmma_f32_16x16x4_f32(
                            false, af, false, bf[ct], (short)0, acc[i][ct], false, false);
                }
            }
        }
    }
    __syncthreads();
    // D -> LDS  (VGPR v holds row v (lanes 0-15) / v+8 (lanes 16-31))
    #pragma unroll
    for (int i = 0; i < 2; ++i) {
        int rt = wave + i*4;
        if (rt < nRT) {
            #pragma unroll
            for (int ct = 0; ct < 3; ++ct)
                #pragma unroll
                for (int v = 0; v < 8; ++v)
                    Dbuf[rt*16 + v + 8*lhalf][ct*16 + l15] = acc[i][ct][v];
        }
    }
    __syncthreads();

    // GEMM2: directions = Wd @ D
    v8f acc2[2][3];
    #pragma unroll
    for (int i = 0; i < 2; ++i)
        #pragma unroll
        for (int ct = 0; ct < 3; ++ct) acc2[i][ct] = (v8f){};

    const int steps2 = Cout >> 2;
    for (int s = 0; s < steps2; ++s) {
        int kg = s*4 + lhalf*2;
        v2f bf[3];
        #pragma unroll
        for (int ct = 0; ct < 3; ++ct) {
            bf[ct].x = Dbuf[kg][ct*16 + l15];
            bf[ct].y = Dbuf[kg+1][ct*16 + l15];
        }
        #pragma unroll
        for (int i = 0; i < 2; ++i) {
            int rt = wave + i*4;
            if (rt < nRT) {
                int row = rt*16 + l15;
                v2f af;
                af.x = Wd[row*Cout + kg];
                af.y = Wd[row*Cout + kg + 1];
                #pragma unroll
                for (int ct = 0; ct < 3; ++ct)
                    acc2[i][ct] = __builtin_amdgcn_wmma_f32_16x16x4_f32(
                        false, af, false, bf[ct], (short)0, acc2[i][ct], false, false);
            }
        }
    }
    #pragma unroll
    for (int i = 0; i < 2; ++i) {
        int rt = wave + i*4;
        if (rt < nRT) {
            #pragma unroll
            for (int ct = 0; ct < 3; ++ct)
                #pragma unroll
                for (int v = 0; v < 8; ++v)
                    D2buf[rt*16 + v + 8*lhalf][ct*16 + l15] = acc2[i][ct][v];
        }
    }
    __syncthreads();

    // VN-LeakyReLU + mean over K, one channel per thread
    if (tid < Cout) {
        float a3[3] = {0.f, 0.f, 0.f};
        for (int k = 0; k < KNN; ++k) {
            float xv[3] = { Dbuf[tid][k],  Dbuf[tid][16+k],  Dbuf[tid][32+k] };
            float dv[3] = { D2buf[tid][k], D2buf[tid][16+k], D2buf[tid][32+k] };
            float o3[3];
            vnact3(xv, dv, o3);
            a3[0] += o3[0]; a3[1] += o3[1]; a3[2] += o3[2];
        }
        const float sc = 1.0f / (float)KNN;
        #pragma unroll
        for (int d = 0; d < 3; ++d)
            Xout[((b*CCAT + coffOut + tid)*3 + d)*NPTS + n] = a3[d] * sc;
    }
}

// ---------------------------------------------------------------------------
// cc layer: y[b,co,d] = mean_n vec_act(ccD, ccW @ xcat[:, :, n]).
// One block = (batch, 16-point tile); WMMA GEMM 128x352 over 48 columns.
// Staging is a pure copy -> use GLOBAL_LOAD_ASYNC_TO_LDS_B32 (ASYNCcnt path).
__global__ __launch_bounds__(128) void cc_kernel(const float* __restrict__ xcat,
                                                 const float* __restrict__ ccW,
                                                 const float* __restrict__ ccD,
                                                 float* __restrict__ yAcc) {
    __shared__ float Es[64][48];
    __shared__ float Dbuf[128][48];
    __shared__ float dL[48];

    const int b   = blockIdx.y;
    const int p0  = blockIdx.x * 16;
    const int tid = threadIdx.x;
    const int wave = tid >> 5, lane = tid & 31;
    const int lhalf = lane >> 4, l15 = lane & 15;
    const int K2 = CCAT;

    v8f acc[2][3];
    #pragma unroll
    for (int i = 0; i < 2; ++i)
        #pragma unroll
        for (int ct = 0; ct < 3; ++ct) acc[i][ct] = (v8f){};

    for (int kt = 0; kt < 6; ++kt) {
        __syncthreads();
        for (int e = tid; e < 64*48; e += 128) {
            int r = e / 48, cidx = e - r*48;
            int d = cidx >> 4, p = cidx & 15;
            int g = kt*64 + r;
            if (g < K2) {
                // async DMA: global -> LDS, 4 bytes per active lane, no VGPR data
                unsigned lds = (unsigned)(unsigned long long)&Es[r][cidx];
                unsigned long long ga =
                    (unsigned long long)&xcat[((b*CCAT + g)*3 + d)*NPTS + p0 + p];
                asm volatile("global_load_async_to_lds_b32 %0, %1, off"
                             :: "v"(lds), "v"(ga) : "memory");
            } else {
                Es[r][cidx] = 0.0f;
            }
        }
        wait_async0();
        __syncthreads();
        for (int s = 0; s < 16; ++s) {
            int kg   = kt*64 + s*4 + lhalf*2;
            int rloc = s*4 + lhalf*2;
            v2f bf[3];
            #pragma unroll
            for (int ct = 0; ct < 3; ++ct) {
                bf[ct].x = Es[rloc][ct*16 + l15];
                bf[ct].y = Es[rloc+1][ct*16 + l15];
            }
            #pragma unroll
            for (int i = 0; i < 2; ++i) {
                int rt = wave + i*4;
                int row = rt*16 + l15;
                v2f af;
                af.x = (kg     < K2) ? ccW[row*K2 + kg]     : 0.0f;
                af.y = (kg + 1 < K2) ? ccW[row*K2 + kg + 1] : 0.0f;
                #pragma unroll
                for (int ct = 0; ct < 3; ++ct)
                    acc[i][ct] = __builtin_amdgcn_wmma_f32_16x16x4_f32(
                        false, af, false, bf[ct], (short)0, acc[i][ct], false, false);
            }
        }
    }
    __syncthreads();
    #pragma unroll
    for (int i = 0; i < 2; ++i) {
        int rt = wave + i*4;
        #pragma unroll
        for (int ct = 0; ct < 3; ++ct)
            #pragma unroll
            for (int v = 0; v < 8; ++v)
                Dbuf[rt*16 + v + 8*lhalf][ct*16 + l15] = acc[i][ct][v];
    }
    __syncthreads();
    if (tid < 48) {  // single shared direction per point: d = ccD . out
        float s = 0.0f;
        for (int c = 0; c < 128; ++c) s += ccD[c] * Dbuf[c][tid];
        dL[tid] = s;
    }
    __syncthreads();
    {
        float a3[3] = {0.f, 0.f, 0.f};
        for (int p = 0; p < 16; ++p) {
            float dv[3] = { dL[p], dL[16+p], dL[32+p] };
            float xv[3] = { Dbuf[tid][p], Dbuf[tid][16+p], Dbuf[tid][32+p] };
            float o3[3];
            vnact3(xv, dv, o3);
            a3[0] += o3[0]; a3[1] += o3[1]; a3[2] += o3[2];
        }
        atomicAdd(&yAcc[(b*128 + tid)*3 + 0], a3[0]);
        atomicAdd(&yAcc[(b*128 + tid)*3 + 1], a3[1]);
        atomicAdd(&yAcc[(b*128 + tid)*3 + 2], a3[2]);
    }
}

// ---------------------------------------------------------------------------
// Head: cevn, polar(SVD) rotation, scale, center MLP, mean/logvar MLPs.
__device__ __forceinline__ void gemv3(const float* __restrict__ W, int ncol,
                                      const float (*src)[3], float out[3], int row) {
    float s0 = 0.f, s1 = 0.f, s2 = 0.f;
    for (int c = 0; c < ncol; ++c) {
        float w = W[row*ncol + c];
        s0 += w*src[c][0]; s1 += w*src[c][1]; s2 += w*src[c][2];
    }
    out[0] = s0; out[1] = s1; out[2] = s2;
}

__global__ __launch_bounds__(128) void head_kernel(
    const float* __restrict__ yAcc, const float* __restrict__ fcO,
    const float* __restrict__ m0W, const float* __restrict__ m0D,
    const float* __restrict__ m1W, const float* __restrict__ m1D,
    const float* __restrict__ l0W, const float* __restrict__ l0D,
    const float* __restrict__ l1W, const float* __restrict__ l1D,
    const float* __restrict__ fimW, const float* __restrict__ filW,
    const float* __restrict__ rbDin, const float* __restrict__ rbW0,
    const float* __restrict__ rbDh, const float* __restrict__ rbW1,
    const float* __restrict__ rbWs, float* __restrict__ out) {

    __shared__ float ys[128][3], so3f[128][3], tA[128][3], tB[128][3];
    __shared__ float nrm[128];
    __shared__ float sc[2];
    __shared__ float Rm[9];

    const int b = blockIdx.x, t = threadIdx.x;
    const float invN = 1.0f / (float)NPTS;

    {   // y = yAcc / N, per-channel norms
        float v0 = yAcc[(b*128 + t)*3 + 0] * invN;
        float v1 = yAcc[(b*128 + t)*3 + 1] * invN;
        float v2 = yAcc[(b*128 + t)*3 + 2] * invN;
        ys[t][0] = v0; ys[t][1] = v1; ys[t][2] = v2;
        nrm[t] = sqrtf(v0*v0 + v1*v1 + v2*v2);
    }
    __syncthreads();
    if (t == 0) {
        float s2 = 0.f, s1 = 0.f;
        for (int c = 0; c < 128; ++c) { s2 += nrm[c]*nrm[c]; s1 += nrm[c]; }
        sc[0] = 1.0f / fmaxf(sqrtf(s2), VEPS);
        out[1060 + b] = (s1 * (1.0f/128.0f) + VEPS) * 640.0f;   // scale
    }
    __syncthreads();
    {   // cevn(y): also equals yn used by mean/logvar paths
        float invc = 1.0f / fmaxf(nrm[t], VEPS);
        float nn = nrm[t] * sc[0];
        so3f[t][0] = ys[t][0]*invc*nn;
        so3f[t][1] = ys[t][1]*invc*nn;
        so3f[t][2] = ys[t][2]*invc*nn;
    }
    __syncthreads();
    if (t < 9) {   // R[d][o] = sum_c fcO[o,c]*so3f[c][d]
        int o = t % 3, dd = t / 3;
        float s = 0.f;
        for (int c = 0; c < 128; ++c) s += fcO[o*128 + c] * so3f[c][dd];
        Rm[dd*3 + o] = s;
    }
    __syncthreads();
    if (t == 0) {   // polar factor of R via Newton: Q <- 0.5*(Q + Q^{-1})
        float Q[9];
        #pragma unroll
        for (int i = 0; i < 9; ++i) Q[i] = Rm[i];
        for (int it = 0; it < 12; ++it) {
            float a = Q[0], bb = Q[1], c = Q[2];
            float d = Q[3], e  = Q[4], f = Q[5];
            float g = Q[6], h  = Q[7], i2 = Q[8];
            float C0 = e*i2 - f*h,  C1 = f*g - d*i2, C2 = d*h - e*g;
            float C3 = c*h - bb*i2, C4 = a*i2 - c*g, C5 = bb*g - a*h;
            float C6 = bb*f - c*e,  C7 = c*d - a*f,  C8 = a*e - bb*d;
            float det = a*C0 + bb*C1 + c*C2;
            float dm  = (fabsf(det) > 1e-20f) ? det : ((det >= 0.f) ? 1e-20f : -1e-20f);
            float inv = 1.0f / dm;
            Q[0] = 0.5f*(Q[0] + C0*inv); Q[1] = 0.5f*(Q[1] + C3*inv); Q[2] = 0.5f*(Q[2] + C6*inv);
            Q[3] = 0.5f*(Q[3] + C1*inv); Q[4] = 0.5f*(Q[4] + C4*inv); Q[5] = 0.5f*(Q[5] + C7*inv);
            Q[6] = 0.5f*(Q[6] + C2*inv); Q[7] = 0.5f*(Q[7] + C5*inv); Q[8] = 0.5f*(Q[8] + C8*inv);
        }
        // so3 = transpose(polar(R))
        for (int i = 0; i < 3; ++i)
            for (int j = 0; j < 3; ++j)
                out[1024 + b*9 + i*3 + j] = Q[j*3 + i];
    }
    __syncthreads();
    {   // center path: vec_act(rbDin, y)
        float dv[3]; gemv3(rbDin, 128, ys, dv, t);
        float xv[3] = { ys[t][0], ys[t][1], ys[t][2] }, o3[3];
        vnact3(xv, dv, o3);
        tA[t][0] = o3[0]; tA[t][1] = o3[1]; tA[t][2] = o3[2];
    }
    __syncthreads();
    if (t < 64) { float s[3]; gemv3(rbW0, 128, tA, s, t); tB[t][0]=s[0]; tB[t][1]=s[1]; tB[t][2]=s[2]; }
    __syncthreads();
    if (t < 64) {   // vec_act(rbDh, net)
        float dv[3]; gemv3(rbDh, 64, tB, dv, t);
        float xv[3] = { tB[t][0], tB[t][1], tB[t][2] }, o3[3];
        vnact3(xv, dv, o3);
        tA[t][0] = o3[0]; tA[t][1] = o3[1]; tA[t][2] = o3[2];
    }
    __syncthreads();
    if (t < 3) {
        float dx = 0.f;
        for (int h = 0; h < 64; ++h) dx += rbW1[h] * tA[h][t];
        float yy = 0.f;
        for (int c = 0; c < 128; ++c) yy += rbWs[c] * ys[c][t];
        out[1064 + b*3 + t] = (yy + dx) * 640.0f;
    }
    __syncthreads();

    // ----- mean path: vlna(m1, vlna(m0, yn)), yn == so3f -----
    { float s[3]; gemv3(m0W, 128, so3f, s, t); tA[t][0]=s[0]; tA[t][1]=s[1]; tA[t][2]=s[2]; }
    __syncthreads();
    { float dv[3]; gemv3(m0D, 128, tA, dv, t);
      float xv[3] = { tA[t][0], tA[t][1], tA[t][2] }, o3[3];
      vnact3(xv, dv, o3); tB[t][0]=o3[0]; tB[t][1]=o3[1]; tB[t][2]=o3[2]; }
    __syncthreads();
    { float s[3]; gemv3(m1W, 128, tB, s, t); tA[t][0]=s[0]; tA[t][1]=s[1]; tA[t][2]=s[2]; }
    __syncthreads();
    { float dv[3]; gemv3(m1D, 128, tA, dv, t);
      float xv[3] = { tA[t][0], tA[t][1], tA[t][2] }, o3[3];
      vnact3(xv, dv, o3); tB[t][0]=o3[0]; tB[t][1]=o3[1]; tB[t][2]=o3[2]; }
    __syncthreads();
    { float g3[3]; gemv3(fimW, 128, tB, g3, t);
      out[b*128 + t] = tB[t][0]*g3[0] + tB[t][1]*g3[1] + tB[t][2]*g3[2]; }
    __syncthreads();

    // ----- logvar path -----
    { float s[3]; gemv3(l0W, 128, so3f, s, t); tA[t][0]=s[0]; tA[t][1]=s[1]; tA[t][2]=s[2]; }
    __syncthreads();
    { float dv[3]; gemv3(l0D, 128, tA, dv, t);
      float xv[3] = { tA[t][0], tA[t][1], tA[t][2] }, o3[3];
      vnact3(xv, dv, o3); tB[t][0]=o3[0]; tB[t][1]=o3[1]; tB[t][2]=o3[2]; }
    __syncthreads();
    { float s[3]; gemv3(l1W, 128, tB, s, t); tA[t][0]=s[0]; tA[t][1]=s[1]; tA[t][2]=s[2]; }
    __syncthreads();
    { float dv[3]; gemv3(l1D, 128, tA, dv, t);
      float xv[3] = { tA[t][0], tA[t][1], tA[t][2] }, o3[3];
      vnact3(xv, dv, o3); tB[t][0]=o3[0]; tB[t][1]=o3[1]; tB[t][2]=o3[2]; }
    __syncthreads();
    { float g3[3]; gemv3(filW, 128, tB, g3, t);
      out[512 + b*128 + t] = tB[t][0]*g3[0] + tB[t][1]*g3[1] + tB[t][2]*g3[2]; }
}

// ---------------------------------------------------------------------------
extern "C" void kernel_launch(void* const* d_in, const int* in_sizes, int n_in,
                              void* d_out, int out_size, void* d_ws, size_t ws_size,
                              hipStream_t stream) {
    (void)in_sizes; (void)n_in; (void)out_size; (void)ws_size;

    const float* x    = (const float*)d_in[0];
    const float* c1W  = (const float*)d_in[2];
    const float* c1D  = (const float*)d_in[3];
    const float* c2W  = (const float*)d_in[4];
    const float* c2D  = (const float*)d_in[5];
    const float* c3W  = (const float*)d_in[6];
    const float* c3D  = (const float*)d_in[7];
    const float* c4W  = (const float*)d_in[8];
    const float* c4D  = (const float*)d_in[9];
    const float* ccW  = (const float*)d_in[10];
    const float* ccD  = (const float*)d_in[11];
    const float* fcO  = (const float*)d_in[12];
    const float* m0W  = (const float*)d_in[13];
    const float* m0D  = (const float*)d_in[14];
    const float* m1W  = (const float*)d_in[15];
    const float* m1D  = (const float*)d_in[16];
    const float* l0W  = (const float*)d_in[17];
    const float* l0D  = (const float*)d_in[18];
    const float* l1W  = (const float*)d_in[19];
    const float* l1D  = (const float*)d_in[20];
    const float* fimW = (const float*)d_in[21];
    const float* filW = (const float*)d_in[22];
    const float* rbDin= (const float*)d_in[23];
    const float* rbW0 = (const float*)d_in[24];
    const float* rbDh = (const float*)d_in[25];
    const float* rbW1 = (const float*)d_in[26];
    const float* rbWs = (const float*)d_in[27];

    char* ws = (char*)d_ws;
    int*   idx  = (int*)ws;                                      // 4*1024*16 ints
    float* xcat = (float*)(ws + (size_t)4*NPTS*KNN*4);           // (4,352,3,1024) f32
    float* yAcc = (float*)(ws + (size_t)4*NPTS*KNN*4
                              + (size_t)4*CCAT*3*NPTS*4);        // (4,128,3) f32
    float* out  = (float*)d_out;

    knn_kernel<<<4, 256, 0, stream>>>(x, idx);
    zero_kernel<<<(4*128*3 + 255)/256, 256, 0, stream>>>(yAcc, 4*128*3);

    // EdgeConv stack; outputs land in the concat buffer at their channel offset
    edgeconv_kernel<<<dim3(NPTS, 4), 128, 0, stream>>>(x,    1,    0,   1, c1W, c1D,  32, idx, xcat,   0);
    edgeconv_kernel<<<dim3(NPTS, 4), 128, 0, stream>>>(xcat, CCAT, 0,  32, c2W, c2D,  64, idx, xcat,  32);
    edgeconv_kernel<<<dim3(NPTS, 4), 128, 0, stream>>>(xcat, CCAT, 32, 64, c3W, c3D, 128, idx, xcat,  96);
    edgeconv_kernel<<<dim3(NPTS, 4), 128, 0, stream>>>(xcat, CCAT, 96,128, c4W, c4D, 128, idx, xcat, 224);

    cc_kernel<<<dim3(NPTS/16, 4), 128, 0, stream>>>(xcat, ccW, ccD, yAcc);

    head_kernel<<<4, 128, 0, stream>>>(yAcc, fcO, m0W, m0D, m1W, m1D,
                                       l0W, l0D, l1W, l1D, fimW, filW,
                                       rbDin, rbW0, rbDh, rbW1, rbWs, out);
}